// GProjection_20048907337781
// MI455X (gfx1250) — compile-verified
//
#include <hip/hip_runtime.h>
#include <hip/hip_bf16.h>
#include <cstdint>

#define PI_F 3.14159265358979323846f
#define DEG_F 0.017453292519943295f

constexpr int B = 8, V = 3, N = 8192;
constexpr int C0 = 64,  H0 = 56;
constexpr int C1 = 128, H1 = 28;
constexpr int C2 = 256, H2 = 14;
constexpr int CH = C0 + C1 + C2;      // 448
constexpr int OUTC = 3 + 3 * CH;      // 1347
constexpr float CAM_F = 248.0f, CAM_C = 111.5f;

constexpr int PPB = 2;                // points per block
constexpr int TPP = CH / 4;           // 112 threads per point, 4 channels each
constexpr int BLK = PPB * TPP;        // 224 = 7 full waves

// ---------------------------------------------------------------------------
// Per-(b,v) camera rotation rows (X,Y,Z normalized) + origin, mirroring the
// reference's _camera_mat computation order exactly.
// cams[i*12 + 0..8] = cm row-major, cams[i*12 + 9..11] = origin.
// ---------------------------------------------------------------------------
__global__ void cam_setup_kernel(const float* __restrict__ poses,
                                 float* __restrict__ cams) {
  int i = threadIdx.x;
  if (i >= B * V) return;
  const float* p = poses + i * 5;
  float theta = p[0] * DEG_F;
  float elr   = p[1] * DEG_F;
  float dist  = p[3];
  float camy  = dist * sinf(elr);
  float lens  = dist * cosf(elr);
  float camx  = lens * cosf(theta);
  float camz  = lens * sinf(theta);
  float Zx = camx, Zy = camy, Zz = camz;
  float tp = theta + PI_F;
  float Yx = camy * cosf(tp), Yy = lens, Yz = camy * sinf(tp);
  float Xx = Yy * Zz - Yz * Zy;
  float Xy = Yz * Zx - Yx * Zz;
  float Xz = Yx * Zy - Yy * Zx;
  float nX = sqrtf(Xx * Xx + Xy * Xy + Xz * Xz);
  float nY = sqrtf(Yx * Yx + Yy * Yy + Yz * Yz);
  float nZ = sqrtf(Zx * Zx + Zy * Zy + Zz * Zz);
  float* c = cams + i * 12;
  c[0] = Xx / nX; c[1] = Xy / nX; c[2] = Xz / nX;
  c[3] = Yx / nY; c[4] = Yy / nY; c[5] = Yz / nY;
  c[6] = Zx / nZ; c[7] = Zy / nZ; c[8] = Zz / nZ;
  c[9] = Zx; c[10] = Zy; c[11] = Zz;
}

// ---------------------------------------------------------------------------
// [vb, C, HW] -> [vb, HW, C] tiled transpose.
// Phase 1 stages a 32x32 tile into LDS via CDNA5 GLOBAL_LOAD_ASYNC_TO_LDS_B32
// (ASYNCcnt-tracked, bypasses VGPRs), then s_wait_asynccnt + barrier.
// Phase 2 writes coalesced channel-last. LDS tile padded (32x33) => no bank
// conflicts in the transposed read.
// ---------------------------------------------------------------------------
__global__ __launch_bounds__(256) void transpose_kernel(
    const float* __restrict__ in, float* __restrict__ out, int C, int HW) {
  __shared__ float tile[32 * 33];
  const int vb = blockIdx.z;
  const int s0 = blockIdx.x * 32;
  const int c0 = blockIdx.y * 32;
  const int t  = threadIdx.x;       // 256 threads
  const int sl = t & 31;            // spatial-in-tile (phase 1)
  const int ch = t >> 5;            // 0..7
  const float* src = in + ((size_t)vb * C + c0) * (size_t)HW;
  const int sidx = min(s0 + sl, HW - 1);   // clamp: duplicate loads never written out
  // LDS byte offset: generic addr low 32 bits are the LDS offset (flat aperture).
  unsigned ldsbase = (unsigned)(uintptr_t)tile;
#pragma unroll
  for (int k = 0; k < 4; ++k) {
    int cc = ch + k * 8;
    unsigned ldsaddr = ldsbase + (unsigned)((cc * 33 + sl) * 4);
    unsigned long long ga =
        (unsigned long long)(uintptr_t)(src + (size_t)cc * HW + sidx);
    asm volatile("global_load_async_to_lds_b32 %0, %1, off"
                 :: "v"(ldsaddr), "v"(ga) : "memory");
  }
  asm volatile("s_wait_asynccnt 0" ::: "memory");
  __syncthreads();
  const int cw = t & 31;            // channel-in-tile (phase 2)
  const int sw = t >> 5;
#pragma unroll
  for (int k = 0; k < 4; ++k) {
    int ss = sw + k * 8;
    if (s0 + ss < HW)
      out[((size_t)vb * HW + (size_t)(s0 + ss)) * C + (c0 + cw)] = tile[cw * 33 + ss];
  }
}

// ---------------------------------------------------------------------------
// Main kernel: 2 points per block, 112 threads/point, 4 channels/thread.
// Threads lt<3 of each point compute the 3 view projections into LDS and copy
// raw xyz to the output; every thread samples 3 views x 4 corners with float4
// (16B-aligned, coalesced channel-last) loads and writes max/mean/std.
// ---------------------------------------------------------------------------
template <int TRANSPOSED>
__global__ __launch_bounds__(BLK) void gproj_main_kernel(
    const float* __restrict__ inp, const int* __restrict__ res,
    const float* __restrict__ cams,
    const float* __restrict__ f0, const float* __restrict__ f1,
    const float* __restrict__ f2, float* __restrict__ out) {
  __shared__ float gxy[PPB][V][2];
  const int t  = threadIdx.x;
  const int lp = (t >= TPP) ? 1 : 0;       // local point
  const int lt = t - lp * TPP;             // thread within point
  const int p  = blockIdx.x * PPB + lp;
  const int b  = p / N;

  if (lt < V) {
    const int v = lt;
    const float* cm0 = cams + (size_t)(b * V) * 12;
    const float* cmv = cams + (size_t)(b * V + v) * 12;
    float px = inp[(size_t)p * 3 + 0];
    float py = inp[(size_t)p * 3 + 1];
    float pz = inp[(size_t)p * 3 + 2] - 0.8f;          // + MESH_POS
    // world = pos @ cm0 + o0   (camera_trans_inv of view 0)
    float wx = px * cm0[0] + py * cm0[3] + pz * cm0[6] + cm0[9];
    float wy = px * cm0[1] + py * cm0[4] + pz * cm0[7] + cm0[10];
    float wz = px * cm0[2] + py * cm0[5] + pz * cm0[8] + cm0[11];
    // q = cm_v @ (world - o_v)  (camera_trans of view v)
    float dx = wx - cmv[9], dy = wy - cmv[10], dz = wz - cmv[11];
    float qx = cmv[0] * dx + cmv[1] * dy + cmv[2] * dz;
    float qy = cmv[3] * dx + cmv[4] * dy + cmv[5] * dz;
    float qz = cmv[6] * dx + cmv[7] * dy + cmv[8] * dz;
    float half0 = ((float)res[0] - 1.0f) * 0.5f;
    float half1 = ((float)res[1] - 1.0f) * 0.5f;
    float coff0 = CAM_C - half0, coff1 = CAM_C - half1;
    float w = (-CAM_F * (qx / qz) + coff0) / half0;
    float h = ( CAM_F * (qy / qz) + coff1) / half1;
    gxy[lp][v][0] = fminf(fmaxf(w, -1.0f), 1.0f);
    gxy[lp][v][1] = fminf(fmaxf(h, -1.0f), 1.0f);
    out[(size_t)p * OUTC + v] = inp[(size_t)p * 3 + v];  // raw xyz channels
  }
  __syncthreads();

  const int c4 = lt * 4;                    // first of 4 channels
  int C, H, cl;
  const float* f;
  if (c4 < C0)           { C = C0; H = H0; cl = c4;            f = f0; }
  else if (c4 < C0 + C1) { C = C1; H = H1; cl = c4 - C0;       f = f1; }
  else                   { C = C2; H = H2; cl = c4 - C0 - C1;  f = f2; }

  float s[V][4];
#pragma unroll
  for (int v = 0; v < V; ++v) {
    float gx = gxy[lp][v][0], gy = gxy[lp][v][1];
    // torch grid_sample, align_corners=False
    float x = ((gx + 1.0f) * (float)H - 1.0f) * 0.5f;
    float y = ((gy + 1.0f) * (float)H - 1.0f) * 0.5f;
    float x0f = floorf(x), y0f = floorf(y);
    float wx1 = x - x0f, wy1 = y - y0f;
    float wx0 = 1.0f - wx1, wy0 = 1.0f - wy1;
    const int vbH = (v * B + b) * H;
    float a0 = 0.f, a1 = 0.f, a2 = 0.f, a3 = 0.f;
#pragma unroll
    for (int j = 0; j < 2; ++j) {
#pragma unroll
      for (int i = 0; i < 2; ++i) {
        float xc = x0f + (float)i, yc = y0f + (float)j;
        bool valid = (xc >= 0.0f) && (xc <= (float)(H - 1)) &&
                     (yc >= 0.0f) && (yc <= (float)(H - 1));
        int xi = (int)fminf(fmaxf(xc, 0.0f), (float)(H - 1));
        int yi = (int)fminf(fmaxf(yc, 0.0f), (float)(H - 1));
        float wgt = (i ? wx1 : wx0) * (j ? wy1 : wy0);
        wgt = valid ? wgt : 0.0f;
        if (TRANSPOSED) {
          const float4 val =
              *(const float4*)(f + (size_t)((vbH + yi) * H + xi) * C + cl);
          a0 = fmaf(val.x, wgt, a0);
          a1 = fmaf(val.y, wgt, a1);
          a2 = fmaf(val.z, wgt, a2);
          a3 = fmaf(val.w, wgt, a3);
        } else {
          const int HW = H * H;
          const float* q =
              f + (size_t)(((v * B + b) * C + cl) * HW) + yi * H + xi;
          a0 = fmaf(q[0 * HW], wgt, a0);
          a1 = fmaf(q[1 * HW], wgt, a1);
          a2 = fmaf(q[2 * HW], wgt, a2);
          a3 = fmaf(q[3 * HW], wgt, a3);
        }
      }
    }
    s[v][0] = a0; s[v][1] = a1; s[v][2] = a2; s[v][3] = a3;
  }

  const size_t row = (size_t)p * OUTC + 3 + (size_t)c4;
#pragma unroll
  for (int k = 0; k < 4; ++k) {
    float va = s[0][k], vb2 = s[1][k], vc = s[2][k];
    float mx   = fmaxf(fmaxf(va, vb2), vc);
    float mean = (va + vb2 + vc) / 3.0f;
    float d0 = va - mean, d1 = vb2 - mean, d2 = vc - mean;
    float sd = sqrtf((d0 * d0 + d1 * d1 + d2 * d2) * 0.5f);  // ddof=1
    out[row + k]          = mx;
    out[row + CH + k]     = mean;
    out[row + 2 * CH + k] = sd;
  }
}

// ---------------------------------------------------------------------------
extern "C" void kernel_launch(void* const* d_in, const int* in_sizes, int n_in,
                              void* d_out, int out_size, void* d_ws, size_t ws_size,
                              hipStream_t stream) {
  (void)in_sizes; (void)n_in; (void)out_size;
  const float* feat0      = (const float*)d_in[0];
  const float* feat1      = (const float*)d_in[1];
  const float* feat2      = (const float*)d_in[2];
  const float* inputs     = (const float*)d_in[3];
  const float* poses      = (const float*)d_in[4];
  const int*   resolution = (const int*)d_in[5];
  float* out = (float*)d_out;
  float* ws  = (float*)d_ws;

  const size_t n0 = (size_t)V * B * C0 * H0 * H0;
  const size_t n1 = (size_t)V * B * C1 * H1 * H1;
  const size_t n2 = (size_t)V * B * C2 * H2 * H2;
  float* cams = ws;                 // 288 floats
  float* fT0  = ws + 288;
  float* fT1  = fT0 + n0;
  float* fT2  = fT1 + n1;
  const size_t need = (288 + n0 + n1 + n2) * sizeof(float);
  const int transposed = (ws_size >= need) ? 1 : 0;

  cam_setup_kernel<<<1, 32, 0, stream>>>(poses, cams);

  if (transposed) {
    dim3 g0((H0 * H0 + 31) / 32, C0 / 32, V * B);
    transpose_kernel<<<g0, 256, 0, stream>>>(feat0, fT0, C0, H0 * H0);
    dim3 g1((H1 * H1 + 31) / 32, C1 / 32, V * B);
    transpose_kernel<<<g1, 256, 0, stream>>>(feat1, fT1, C1, H1 * H1);
    dim3 g2((H2 * H2 + 31) / 32, C2 / 32, V * B);
    transpose_kernel<<<g2, 256, 0, stream>>>(feat2, fT2, C2, H2 * H2);
    gproj_main_kernel<1><<<(B * N) / PPB, BLK, 0, stream>>>(
        inputs, resolution, cams, fT0, fT1, fT2, out);
  } else {
    gproj_main_kernel<0><<<(B * N) / PPB, BLK, 0, stream>>>(
        inputs, resolution, cams, feat0, feat1, feat2, out);
  }
}